// AttnGCN_62629213110802
// MI455X (gfx1250) — compile-verified
//
#include <hip/hip_runtime.h>

// ---------------- problem constants (match reference) ----------------
static constexpr int N = 100000;
static constexpr int E = 1600000;
static constexpr int ETOT = E + N;          // with self loops
static constexpr float EPS_BN = 1e-5f;
static constexpr float NEG_SLOPE = 0.2f;

typedef __attribute__((ext_vector_type(2))) float v2f;
typedef __attribute__((ext_vector_type(8))) float v8f;

static inline int cdiv(long long a, long long b) { return (int)((a + b - 1) / b); }

// ---------------- float atomic max via int/uint monotonic trick ----------------
__device__ __forceinline__ void atomicMaxF(float* addr, float val) {
    if (val >= 0.0f) atomicMax((int*)addr, __float_as_int(val));
    else             atomicMin((unsigned int*)addr, __float_as_uint(val));
}

// ---------------- BN statistics: block reduce + atomics ----------------
__global__ __launch_bounds__(256) void bnStatsKernel(const float* __restrict__ h,
                                                     float* __restrict__ stats) {
    __shared__ float s0[256], s1[256], s2[256], s3[256];
    float a0 = 0.f, a1 = 0.f, a2 = 0.f, a3 = 0.f;
    for (int i = blockIdx.x * blockDim.x + threadIdx.x; i < N; i += gridDim.x * blockDim.x) {
        float x0 = h[2 * i], x1 = h[2 * i + 1];
        a0 += x0; a1 += x1; a2 += x0 * x0; a3 += x1 * x1;
    }
    int t = threadIdx.x;
    s0[t] = a0; s1[t] = a1; s2[t] = a2; s3[t] = a3;
    __syncthreads();
    for (int s = 128; s > 0; s >>= 1) {
        if (t < s) { s0[t] += s0[t + s]; s1[t] += s1[t + s]; s2[t] += s2[t + s]; s3[t] += s3[t + s]; }
        __syncthreads();
    }
    if (t == 0) {
        atomicAdd(&stats[0], s0[0]); atomicAdd(&stats[1], s1[0]);
        atomicAdd(&stats[2], s2[0]); atomicAdd(&stats[3], s3[0]);
    }
}

// ---------------- BN + GAT1 projection (Fin=2) + attention logits ----------------
__global__ __launch_bounds__(256) void bnProjKernel(const float* __restrict__ h,
                                                    const float* __restrict__ stats,
                                                    const float* __restrict__ gamma,
                                                    const float* __restrict__ beta,
                                                    const float* __restrict__ W1,   // [64,2]
                                                    const float* __restrict__ as1,  // [64]
                                                    const float* __restrict__ ad1,  // [64]
                                                    float* __restrict__ xp,         // [N,64]
                                                    float* __restrict__ es,         // [N,2]
                                                    float* __restrict__ ed) {       // [N,2]
    int n = blockIdx.x * blockDim.x + threadIdx.x;
    if (n >= N) return;
    float inv = 1.0f / (float)N;
    float mu0 = stats[0] * inv, mu1 = stats[1] * inv;
    float var0 = stats[2] * inv - mu0 * mu0;
    float var1 = stats[3] * inv - mu1 * mu1;
    float x0 = (h[2 * n]     - mu0) * rsqrtf(var0 + EPS_BN) * gamma[0] + beta[0];
    float x1 = (h[2 * n + 1] - mu1) * rsqrtf(var1 + EPS_BN) * gamma[1] + beta[1];
    float es0 = 0.f, es1 = 0.f, ed0 = 0.f, ed1 = 0.f;
#pragma unroll
    for (int o = 0; o < 64; ++o) {
        float v = x0 * W1[2 * o] + x1 * W1[2 * o + 1];
        xp[(size_t)n * 64 + o] = v;
        if (o < 32) { es0 += v * as1[o]; ed0 += v * ad1[o]; }
        else        { es1 += v * as1[o]; ed1 += v * ad1[o]; }
    }
    es[2 * n] = es0; es[2 * n + 1] = es1;
    ed[2 * n] = ed0; ed[2 * n + 1] = ed1;
}

// ---------------- attention logits from xp [N,64] (layer 2) ----------------
__global__ __launch_bounds__(256) void esedKernel(const float* __restrict__ xp,
                                                  const float* __restrict__ a_s,  // [64]
                                                  const float* __restrict__ a_d,  // [64]
                                                  float* __restrict__ es,
                                                  float* __restrict__ ed) {
    int n = blockIdx.x * blockDim.x + threadIdx.x;
    if (n >= N) return;
    float es0 = 0.f, es1 = 0.f, ed0 = 0.f, ed1 = 0.f;
    const float* row = xp + (size_t)n * 64;
#pragma unroll
    for (int o = 0; o < 64; ++o) {
        float v = row[o];
        if (o < 32) { es0 += v * a_s[o]; ed0 += v * a_d[o]; }
        else        { es1 += v * a_s[o]; ed1 += v * a_d[o]; }
    }
    es[2 * n] = es0; es[2 * n + 1] = es1;
    ed[2 * n] = ed0; ed[2 * n + 1] = ed1;
}

// ---------------- init m = -inf, denom = 0 ----------------
__global__ __launch_bounds__(256) void initMDKernel(float* __restrict__ m, float* __restrict__ dn) {
    int i = blockIdx.x * blockDim.x + threadIdx.x;
    if (i < 2 * N) { m[i] = -__builtin_inff(); dn[i] = 0.0f; }
}

// ---------------- edge pass 1: segment max ----------------
__global__ __launch_bounds__(256) void edgeMaxKernel(const int* __restrict__ ei,
                                                     const float* __restrict__ es,
                                                     const float* __restrict__ ed,
                                                     float* __restrict__ m) {
    int t = blockIdx.x * blockDim.x + threadIdx.x;
    if (t >= ETOT * 2) return;
    int eid = t >> 1, hh = t & 1;
    int s, d;
    if (eid < E) { s = ei[eid]; d = ei[E + eid]; } else { s = d = eid - E; }
    float e = es[2 * s + hh] + ed[2 * d + hh];
    e = e > 0.f ? e : NEG_SLOPE * e;
    atomicMaxF(&m[2 * d + hh], e);
}

// ---------------- edge pass 2: segment sum of exp ----------------
__global__ __launch_bounds__(256) void edgeSumKernel(const int* __restrict__ ei,
                                                     const float* __restrict__ es,
                                                     const float* __restrict__ ed,
                                                     const float* __restrict__ m,
                                                     float* __restrict__ dn) {
    int t = blockIdx.x * blockDim.x + threadIdx.x;
    if (t >= ETOT * 2) return;
    int eid = t >> 1, hh = t & 1;
    int s, d;
    if (eid < E) { s = ei[eid]; d = ei[E + eid]; } else { s = d = eid - E; }
    float e = es[2 * s + hh] + ed[2 * d + hh];
    e = e > 0.f ? e : NEG_SLOPE * e;
    atomicAdd(&dn[2 * d + hh], __expf(e - m[2 * d + hh]));
}

// ---------------- edge pass 3: weighted aggregation (64 ch per edge) ----------------
__global__ __launch_bounds__(256) void edgeAggKernel(const int* __restrict__ ei,
                                                     const float* __restrict__ xp,
                                                     const float* __restrict__ es,
                                                     const float* __restrict__ ed,
                                                     const float* __restrict__ m,
                                                     const float* __restrict__ dn,
                                                     float* __restrict__ out) {
    int t = blockIdx.x * blockDim.x + threadIdx.x;
    if (t >= ETOT * 64) return;
    int eid = t >> 6, c = t & 63, hh = c >> 5;
    int s, d;
    if (eid < E) { s = ei[eid]; d = ei[E + eid]; } else { s = d = eid - E; }
    float e = es[2 * s + hh] + ed[2 * d + hh];
    e = e > 0.f ? e : NEG_SLOPE * e;
    float alpha = __expf(e - m[2 * d + hh]) / (dn[2 * d + hh] + 1e-16f);
    atomicAdd(&out[(size_t)d * 64 + c], alpha * xp[(size_t)s * 64 + c]);
}

// ---------------- fp32 WMMA GEMM: OUT[N,outDim] = actA(X[N,64]) @ W[outDim,64]^T ----
// One wave32 per 16x16 output tile; K=64 as 16 chained V_WMMA_F32_16X16X4_F32.
// All behavior flags are compile-time so the K loop is branch-free (no EXEC games).
template <bool RELUA, bool RELUOUT, bool FULLB, bool HASBB>
__global__ __launch_bounds__(256) void gemm64WmmaKernel(const float* __restrict__ X,
                                                        const float* __restrict__ W,
                                                        float* __restrict__ OUT,
                                                        int outDim, int ldo,
                                                        const float* __restrict__ abias,
                                                        const float* __restrict__ bbias,
                                                        int tilesN, int totalTiles) {
    int wave = blockIdx.x * (blockDim.x >> 5) + (threadIdx.x >> 5);
    if (wave >= totalTiles) return;              // wave-uniform: EXEC stays all-1s below
    int tm = wave / tilesN;
    int tn = wave - tm * tilesN;
    int lane = threadIdx.x & 31;
    int laneLow = lane & 15;
    bool hi = lane >= 16;
    int row0 = tm * 16, col0 = tn * 16;
    const float* xrow = X + (size_t)(row0 + laneLow) * 64;
    int bcol = col0 + laneLow;
    bool bvalid = FULLB || (bcol < outDim);
    const float* wrow = W + (size_t)(bvalid ? bcol : 0) * 64;

    v8f acc = {};
#pragma unroll
    for (int k0 = 0; k0 < 64; k0 += 4) {
        int ka = k0 + (hi ? 2 : 0);
        v2f a = *(const v2f*)(xrow + ka);            // 8B-aligned fragment load
        if (RELUA) {
            v2f ab = *(const v2f*)(abias + ka);
            a.x += ab.x; a.y += ab.y;
            a.x = a.x > 0.f ? a.x : 0.f;
            a.y = a.y > 0.f ? a.y : 0.f;
        }
        v2f b = *(const v2f*)(wrow + ka);
        if (!FULLB) {                                 // branchless zero for padded cols
            b.x = bvalid ? b.x : 0.f;
            b.y = bvalid ? b.y : 0.f;
        }
        // D = A(16x4 f32) x B(4x16 f32) + C  -> v_wmma_f32_16x16x4_f32
        acc = __builtin_amdgcn_wmma_f32_16x16x4_f32(
            /*neg_a=*/false, a, /*neg_b=*/false, b,
            /*c_mod=*/(short)0, acc, /*reuse_a=*/false, /*reuse_b=*/false);
    }

    int c = col0 + laneLow;
    if (FULLB || c < outDim) {
        float bb = HASBB ? bbias[c] : 0.0f;
        int rbase = row0 + (hi ? 8 : 0);
#pragma unroll
        for (int v = 0; v < 8; ++v) {
            float val = acc[v] + bb;
            if (RELUOUT) val = val > 0.f ? val : 0.f;
            OUT[(size_t)(rbase + v) * ldo + c] = val;
        }
    }
}

// ---------------------------------------------------------------------------
extern "C" void kernel_launch(void* const* d_in, const int* in_sizes, int n_in,
                              void* d_out, int out_size, void* d_ws, size_t ws_size,
                              hipStream_t stream) {
    (void)in_sizes; (void)n_in; (void)out_size; (void)ws_size;
    const float* h      = (const float*)d_in[0];
    const int*   ei     = (const int*)  d_in[1];
    const float* gamma  = (const float*)d_in[2];
    const float* beta   = (const float*)d_in[3];
    const float* W1     = (const float*)d_in[4];
    const float* as1    = (const float*)d_in[5];
    const float* ad1    = (const float*)d_in[6];
    const float* b1     = (const float*)d_in[7];
    const float* W2     = (const float*)d_in[8];
    const float* as2    = (const float*)d_in[9];
    const float* ad2    = (const float*)d_in[10];
    const float* b2     = (const float*)d_in[11];
    const float* fw1 = (const float*)d_in[12]; const float* fb1 = (const float*)d_in[13];
    const float* fw2 = (const float*)d_in[14]; const float* fb2 = (const float*)d_in[15];
    const float* fw3 = (const float*)d_in[16]; const float* fb3 = (const float*)d_in[17];
    const float* fw4 = (const float*)d_in[18]; const float* fb4 = (const float*)d_in[19];
    const float* fw5 = (const float*)d_in[20]; const float* fb5 = (const float*)d_in[21];
    float* out = (float*)d_out;

    // workspace layout (floats)
    float* ws    = (float*)d_ws;
    float* stats = ws;                       // 16
    float* bufA  = stats + 16;               // N*64  (xp1, then xp2)
    float* bufB  = bufA + (size_t)N * 64;    // N*64  (aggregation accumulator)
    float* bufC  = bufB + (size_t)N * 64;    // N*64  (FFN ping)
    float* bufD  = bufC + (size_t)N * 64;    // N*64  (FFN pong)
    float* es    = bufD + (size_t)N * 64;    // N*2
    float* ed    = es + (size_t)N * 2;       // N*2
    float* mbuf  = ed + (size_t)N * 2;       // N*2
    float* dnbuf = mbuf + (size_t)N * 2;     // N*2

    const int B = 256;
    const int nodeBlocks  = cdiv(N, B);
    const int mdBlocks    = cdiv(2 * N, B);
    const int edge2Blocks = cdiv((long long)ETOT * 2, B);
    const int aggBlocks   = cdiv((long long)ETOT * 64, B);
    const int tilesM = N / 16;                        // 6250
    const int tiles64  = tilesM * 4;                  // outDim 64
    const int g64Blocks = cdiv(tiles64, 8);
    const int tiles16  = tilesM * 1;                  // outDim 11
    const int g16Blocks = cdiv(tiles16, 8);

    // ---- BatchNorm statistics ----
    hipMemsetAsync(stats, 0, 16 * sizeof(float), stream);
    bnStatsKernel<<<256, B, 0, stream>>>(h, stats);

    // ---- BN + GAT1 projection + logits ----
    bnProjKernel<<<nodeBlocks, B, 0, stream>>>(h, stats, gamma, beta, W1, as1, ad1,
                                               bufA, es, ed);

    // ---- GAT1 edge softmax + aggregation ----
    initMDKernel<<<mdBlocks, B, 0, stream>>>(mbuf, dnbuf);
    edgeMaxKernel<<<edge2Blocks, B, 0, stream>>>(ei, es, ed, mbuf);
    edgeSumKernel<<<edge2Blocks, B, 0, stream>>>(ei, es, ed, mbuf, dnbuf);
    hipMemsetAsync(bufB, 0, (size_t)N * 64 * sizeof(float), stream);
    edgeAggKernel<<<aggBlocks, B, 0, stream>>>(ei, bufA, es, ed, mbuf, dnbuf, bufB);

    // ---- GAT2 projection: xp2 = relu(agg1 + b1) @ W2^T  (fp32 WMMA) ----
    gemm64WmmaKernel<true, false, true, false>
        <<<g64Blocks, B, 0, stream>>>(bufB, W2, bufA, 64, 64, b1, nullptr, 4, tiles64);
    esedKernel<<<nodeBlocks, B, 0, stream>>>(bufA, as2, ad2, es, ed);

    // ---- GAT2 edge softmax + aggregation ----
    initMDKernel<<<mdBlocks, B, 0, stream>>>(mbuf, dnbuf);
    edgeMaxKernel<<<edge2Blocks, B, 0, stream>>>(ei, es, ed, mbuf);
    edgeSumKernel<<<edge2Blocks, B, 0, stream>>>(ei, es, ed, mbuf, dnbuf);
    hipMemsetAsync(bufB, 0, (size_t)N * 64 * sizeof(float), stream);
    edgeAggKernel<<<aggBlocks, B, 0, stream>>>(ei, bufA, es, ed, mbuf, dnbuf, bufB);

    // ---- FFN head (all fp32 WMMA GEMMs, bias+ReLU fused) ----
    gemm64WmmaKernel<true, true, true, true>
        <<<g64Blocks, B, 0, stream>>>(bufB, fw1, bufC, 64, 64, b2, fb1, 4, tiles64);
    gemm64WmmaKernel<false, true, true, true>
        <<<g64Blocks, B, 0, stream>>>(bufC, fw2, bufD, 64, 64, nullptr, fb2, 4, tiles64);
    gemm64WmmaKernel<false, true, true, true>
        <<<g64Blocks, B, 0, stream>>>(bufD, fw3, bufC, 64, 64, nullptr, fb3, 4, tiles64);
    gemm64WmmaKernel<false, true, true, true>
        <<<g64Blocks, B, 0, stream>>>(bufC, fw4, bufD, 64, 64, nullptr, fb4, 4, tiles64);
    // final: [N,64] @ fw5[11,64]^T + fb5 (no relu) -> d_out [N,11]
    gemm64WmmaKernel<false, false, false, true>
        <<<g16Blocks, B, 0, stream>>>(bufD, fw5, out, 11, 11, nullptr, fb5, 1, tiles16);
}